// PointNet2_82815559402256
// MI455X (gfx1250) — compile-verified
//
#include <hip/hip_runtime.h>
#include <cfloat>
#include <cstdint>
#include <cstddef>

typedef float v2f __attribute__((ext_vector_type(2)));
typedef float v8f __attribute__((ext_vector_type(8)));

#define BN_EPS 1e-5f

// ---------------------------------------------------------------------------
// pc (B,9,N) -> x0 (B,N,9) and xyz0 (B,N,3)
// ---------------------------------------------------------------------------
__global__ void transpose_pc_kernel(const float* __restrict__ pc,
                                    float* __restrict__ x0,
                                    float* __restrict__ xyz0,
                                    int C, int N, int total) {
  int t = blockIdx.x * blockDim.x + threadIdx.x;
  if (t >= total) return;
  int n = t % N;
  int bc = t / N;
  int c = bc % C;
  int b = bc / C;
  float v = pc[t];
  x0[((size_t)b * N + n) * C + c] = v;
  if (c < 3) xyz0[((size_t)b * N + n) * 3 + c] = v;
}

// ---------------------------------------------------------------------------
// Farthest point sampling: one block per batch, dist[] lives in LDS (320KB/WGP
// budget makes the whole N<=4096 distance array resident).
// Matches jax scan: out[it] = incoming 'far'; argmax ties -> lowest index.
// ---------------------------------------------------------------------------
__global__ void fps_kernel(const float* __restrict__ xyz, int* __restrict__ out,
                           int N, int npoint) {
  int b = blockIdx.x;
  const float* p = xyz + (size_t)b * N * 3;
  extern __shared__ float smem[];
  float* dist = smem;                       // N floats
  float* rv = smem + N;                     // blockDim floats
  int* ri = (int*)(rv + blockDim.x);        // blockDim ints
  __shared__ int far_s;
  for (int i = threadIdx.x; i < N; i += blockDim.x) dist[i] = 1e10f;
  if (threadIdx.x == 0) far_s = 0;
  __syncthreads();
  for (int it = 0; it < npoint; ++it) {
    int f = far_s;
    if (threadIdx.x == 0) out[b * npoint + it] = f;
    float cx = p[(size_t)f * 3 + 0];
    float cy = p[(size_t)f * 3 + 1];
    float cz = p[(size_t)f * 3 + 2];
    float best = -1.0f;
    int bi = 0x7fffffff;
    for (int i = threadIdx.x; i < N; i += blockDim.x) {
      float dx = p[(size_t)i * 3 + 0] - cx;
      float dy = p[(size_t)i * 3 + 1] - cy;
      float dz = p[(size_t)i * 3 + 2] - cz;
      float d = dx * dx + dy * dy + dz * dz;
      float nd = fminf(dist[i], d);
      dist[i] = nd;
      if (nd > best) { best = nd; bi = i; }   // ascending i -> first max kept
    }
    rv[threadIdx.x] = best;
    ri[threadIdx.x] = bi;
    __syncthreads();
    for (int off = blockDim.x >> 1; off > 0; off >>= 1) {
      if (threadIdx.x < off) {
        float v2 = rv[threadIdx.x + off];
        int j2 = ri[threadIdx.x + off];
        float v1 = rv[threadIdx.x];
        int j1 = ri[threadIdx.x];
        if (v2 > v1 || (v2 == v1 && j2 < j1)) {
          rv[threadIdx.x] = v2;
          ri[threadIdx.x] = j2;
        }
      }
      __syncthreads();
    }
    if (threadIdx.x == 0) far_s = ri[0];
    __syncthreads();
  }
}

// new_xyz[b,s,:] = xyz[b, fi[b,s], :]
__global__ void gather_xyz2_kernel(const float* __restrict__ xyz,
                                   const int* __restrict__ fi,
                                   float* __restrict__ nxyz, int N, int S,
                                   int total) {
  int t = blockIdx.x * blockDim.x + threadIdx.x;
  if (t >= total) return;            // total = B*S
  int b = t / S;
  int j = fi[t];
  const float* src = xyz + ((size_t)b * N + j) * 3;
  float* dst = nxyz + (size_t)t * 3;
  dst[0] = src[0];
  dst[1] = src[1];
  dst[2] = src[2];
}

// ---------------------------------------------------------------------------
// Ball query: first K indices (ascending) with d^2 <= r^2; pad with first.
// ---------------------------------------------------------------------------
__global__ void ball_query_kernel(const float* __restrict__ xyz,
                                  const float* __restrict__ nxyz,
                                  int* __restrict__ gi, int N, int S, float r2,
                                  int K, int total) {
  int t = blockIdx.x * blockDim.x + threadIdx.x;
  if (t >= total) return;            // total = B*S
  int b = t / S;
  const float* p = xyz + (size_t)b * N * 3;
  float cx = nxyz[(size_t)t * 3 + 0];
  float cy = nxyz[(size_t)t * 3 + 1];
  float cz = nxyz[(size_t)t * 3 + 2];
  int* out = gi + (size_t)t * K;
  int cnt = 0;
  int first = 0;
  for (int j = 0; j < N; ++j) {
    float dx = p[(size_t)j * 3 + 0] - cx;
    float dy = p[(size_t)j * 3 + 1] - cy;
    float dz = p[(size_t)j * 3 + 2] - cz;
    float d = dx * dx + dy * dy + dz * dz;
    if (d <= r2) {
      if (cnt == 0) first = j;
      out[cnt++] = j;
      if (cnt == K) break;
    }
  }
  for (; cnt < K; ++cnt) out[cnt] = first;
}

// feat row = [xyz[gi]-new_xyz (3) | pts[gi] (C)]
__global__ void group_feat_kernel(const float* __restrict__ xyz,
                                  const float* __restrict__ nxyz,
                                  const float* __restrict__ pts,
                                  const int* __restrict__ gi,
                                  float* __restrict__ X, int N, int S, int K,
                                  int C, int total) {
  int t = blockIdx.x * blockDim.x + threadIdx.x;
  if (t >= total) return;            // total = B*S*K
  int bs = t / K;                    // b*S + s
  int b = bs / S;
  int j = gi[t];
  const float* src = xyz + ((size_t)b * N + j) * 3;
  const float* ctr = nxyz + (size_t)bs * 3;
  float* dst = X + (size_t)t * (3 + C);
  dst[0] = src[0] - ctr[0];
  dst[1] = src[1] - ctr[1];
  dst[2] = src[2] - ctr[2];
  const float* ps = pts + ((size_t)b * N + j) * C;
  for (int c = 0; c < C; ++c) dst[3 + c] = ps[c];
}

// ---------------------------------------------------------------------------
// WMMA fp32 GEMM: Y(R,Cout) = X(R,Cin) @ W(Cout,Cin)^T + bias.
// One wave computes a 16x32 output strip (two 16x16 tiles / two accumulators)
// so every A fragment load feeds two V_WMMA_F32_16X16X4_F32 ops.
// A frag (16x4): lanes 0-15 M=0..15; v0 holds K={0|2}, v1 holds K={1|3}.
// B frag (4x16): same K split with N=lane%16. C/D: v[r] = row r + 8*(lane/16).
// All hot-loop loads are unconditional: out-of-range output columns only feed
// discarded columns of the fragment product, so W row pointers are merely
// clamped to a valid (finite) row; the K remainder zeroes A via arithmetic
// select (no EXEC manipulation anywhere near the WMMAs).
// ---------------------------------------------------------------------------
__global__ void wmma_gemm_bias_kernel(const float* __restrict__ X,
                                      const float* __restrict__ W,
                                      const float* __restrict__ bias,
                                      float* __restrict__ Y, int R, int Cin,
                                      int Cout) {
  int wavesPerBlock = blockDim.x >> 5;
  int wave = blockIdx.x * wavesPerBlock + (threadIdx.x >> 5);
  int lane = threadIdx.x & 31;
  int tilesM = R >> 4;
  int pairsN = (Cout + 31) >> 5;
  if (wave >= tilesM * pairsN) return;   // wave-uniform exit, EXEC stays full
  int tm = wave / pairsN;
  int tp = wave - tm * pairsN;
  int row0 = tm << 4;
  int col0 = tp << 5;                    // strip covers columns col0..col0+31
  int half = lane >> 4;
  int mr = lane & 15;
  int kb = half << 1;                    // 0 or 2
  const float* xr = X + (size_t)(row0 + mr) * Cin + kb;
  int bc0 = col0 + mr;
  int bc1 = col0 + 16 + mr;
  bool v0 = bc0 < Cout;
  bool v1 = bc1 < Cout;
  const float* wr0 = W + (size_t)(v0 ? bc0 : (Cout - 1)) * Cin + kb;
  const float* wr1 = W + (size_t)(v1 ? bc1 : (Cout - 1)) * Cin + kb;
  v8f acc0 = {0.f, 0.f, 0.f, 0.f, 0.f, 0.f, 0.f, 0.f};
  v8f acc1 = {0.f, 0.f, 0.f, 0.f, 0.f, 0.f, 0.f, 0.f};
  int k = 0;
  for (; k + 4 <= Cin; k += 4) {
    v2f a, b0, b1;
    a.x = xr[0];
    a.y = xr[1];
    b0.x = wr0[0];
    b0.y = wr0[1];
    b1.x = wr1[0];
    b1.y = wr1[1];
    acc0 = __builtin_amdgcn_wmma_f32_16x16x4_f32(false, a, false, b0, (short)0,
                                                 acc0, false, false);
    acc1 = __builtin_amdgcn_wmma_f32_16x16x4_f32(false, a, false, b1, (short)0,
                                                 acc1, false, false);
    xr += 4;
    wr0 += 4;
    wr1 += 4;
  }
  if (k < Cin) {                         // remainder 1..3, zero A (not B)
    int k0 = k + kb;
    int k1 = k + kb + 1;
    float m0 = (k0 < Cin) ? 1.f : 0.f;   // arithmetic select -> v_cndmask
    float m1 = (k1 < Cin) ? 1.f : 0.f;
    int o0 = (k0 < Cin) ? 0 : -kb;       // clamp offsets into valid memory
    int o1 = (k1 < Cin) ? 1 : -kb;
    v2f a, b0, b1;
    a.x = xr[o0] * m0;
    a.y = xr[o1] * m1;
    b0.x = wr0[o0];
    b0.y = wr0[o1];
    b1.x = wr1[o0];
    b1.y = wr1[o1];
    acc0 = __builtin_amdgcn_wmma_f32_16x16x4_f32(false, a, false, b0, (short)0,
                                                 acc0, false, false);
    acc1 = __builtin_amdgcn_wmma_f32_16x16x4_f32(false, a, false, b1, (short)0,
                                                 acc1, false, false);
  }
  if (v0) {
    float bv = bias[bc0];
#pragma unroll
    for (int r = 0; r < 8; ++r) {
      Y[(size_t)(row0 + r + (half << 3)) * Cout + bc0] = acc0[r] + bv;
    }
  }
  if (v1) {
    float bv = bias[bc1];
#pragma unroll
    for (int r = 0; r < 8; ++r) {
      Y[(size_t)(row0 + r + (half << 3)) * Cout + bc1] = acc1[r] + bv;
    }
  }
}

// ---------------------------------------------------------------------------
// BN stats: one block per channel; double accumulation for 262k-row sums.
// Writes scale = g*rsqrt(var+eps), shift = bt - mean*scale.
// ---------------------------------------------------------------------------
__global__ void bn_stats_kernel(const float* __restrict__ Y,
                                const float* __restrict__ g,
                                const float* __restrict__ bt,
                                float* __restrict__ scale,
                                float* __restrict__ shift, int R, int C) {
  int c = blockIdx.x;
  double s = 0.0, s2 = 0.0;
  for (int r = threadIdx.x; r < R; r += blockDim.x) {
    float v = Y[(size_t)r * C + c];
    s += (double)v;
    s2 += (double)v * (double)v;
  }
  __shared__ double sh[256];
  __shared__ double sh2[256];
  sh[threadIdx.x] = s;
  sh2[threadIdx.x] = s2;
  __syncthreads();
  for (int off = blockDim.x >> 1; off > 0; off >>= 1) {
    if (threadIdx.x < off) {
      sh[threadIdx.x] += sh[threadIdx.x + off];
      sh2[threadIdx.x] += sh2[threadIdx.x + off];
    }
    __syncthreads();
  }
  if (threadIdx.x == 0) {
    double mean = sh[0] / (double)R;
    double var = sh2[0] / (double)R - mean * mean;
    float sc = g[c] * rsqrtf((float)var + BN_EPS);
    scale[c] = sc;
    shift[c] = bt[c] - (float)mean * sc;
  }
}

__global__ void bn_apply_kernel(float* __restrict__ Y,
                                const float* __restrict__ scale,
                                const float* __restrict__ shift, size_t total,
                                int C) {
  size_t t = (size_t)blockIdx.x * blockDim.x + threadIdx.x;
  if (t >= total) return;
  int c = (int)(t % (size_t)C);
  float v = Y[t] * scale[c] + shift[c];
  Y[t] = v > 0.f ? v : 0.f;
}

// max over K neighbors: X(BS,K,C) -> out(BS,C)
__global__ void maxpool_kernel(const float* __restrict__ X,
                               float* __restrict__ out, int K, int C,
                               int total) {
  int t = blockIdx.x * blockDim.x + threadIdx.x;
  if (t >= total) return;            // total = BS*C
  int c = t % C;
  int bs = t / C;
  const float* p = X + ((size_t)bs * K) * C + c;
  float m = -FLT_MAX;
  for (int k = 0; k < K; ++k) m = fmaxf(m, p[(size_t)k * C]);
  out[t] = m;
}

// copy points1 into first C1 columns of concat buffer
__global__ void copy_cols_kernel(const float* __restrict__ src,
                                 float* __restrict__ dst, int C1, int Ctot,
                                 int total) {
  int t = blockIdx.x * blockDim.x + threadIdx.x;
  if (t >= total) return;            // total = rows*C1
  int c = t % C1;
  int r = t / C1;
  dst[(size_t)r * Ctot + c] = src[t];
}

// ---------------------------------------------------------------------------
// 3-NN inverse-distance interpolation (feature propagation); writes into the
// concat buffer at column offset coff.
// ---------------------------------------------------------------------------
__global__ void fp_interp_kernel(const float* __restrict__ xyz1,
                                 const float* __restrict__ xyz2,
                                 const float* __restrict__ pts2,
                                 float* __restrict__ X, int N1, int N2, int C2,
                                 int Ctot, int coff, int total) {
  int t = blockIdx.x * blockDim.x + threadIdx.x;
  if (t >= total) return;            // total = B*N1
  int b = t / N1;
  float px = xyz1[(size_t)t * 3 + 0];
  float py = xyz1[(size_t)t * 3 + 1];
  float pz = xyz1[(size_t)t * 3 + 2];
  const float* q = xyz2 + (size_t)b * N2 * 3;
  float d0 = FLT_MAX, d1 = FLT_MAX, d2 = FLT_MAX;
  int i0 = 0, i1 = 0, i2 = 0;
  for (int j = 0; j < N2; ++j) {
    float dx = q[(size_t)j * 3 + 0] - px;
    float dy = q[(size_t)j * 3 + 1] - py;
    float dz = q[(size_t)j * 3 + 2] - pz;
    float d = dx * dx + dy * dy + dz * dz;
    if (d < d0) { d2 = d1; i2 = i1; d1 = d0; i1 = i0; d0 = d; i0 = j; }
    else if (d < d1) { d2 = d1; i2 = i1; d1 = d; i1 = j; }
    else if (d < d2) { d2 = d; i2 = j; }
  }
  float w0 = 1.f / (d0 + 1e-8f);
  float w1 = 1.f / (d1 + 1e-8f);
  float w2 = 1.f / (d2 + 1e-8f);
  float ws = w0 + w1 + w2;
  w0 /= ws; w1 /= ws; w2 /= ws;
  const float* p0 = pts2 + ((size_t)b * N2 + i0) * C2;
  const float* p1 = pts2 + ((size_t)b * N2 + i1) * C2;
  const float* p2 = pts2 + ((size_t)b * N2 + i2) * C2;
  float* dst = X + (size_t)t * Ctot + coff;
  for (int c = 0; c < C2; ++c) dst[c] = w0 * p0[c] + w1 * p1[c] + w2 * p2[c];
}

__global__ void log_softmax_kernel(const float* __restrict__ Z,
                                   float* __restrict__ out, int C, int total) {
  int t = blockIdx.x * blockDim.x + threadIdx.x;
  if (t >= total) return;            // total = rows
  const float* z = Z + (size_t)t * C;
  float m = -FLT_MAX;
  for (int c = 0; c < C; ++c) m = fmaxf(m, z[c]);
  float s = 0.f;
  for (int c = 0; c < C; ++c) s += expf(z[c] - m);
  float lse = m + logf(s);
  float* o = out + (size_t)t * C;
  for (int c = 0; c < C; ++c) o[c] = z[c] - lse;
}

// l4p (B,S,C) -> out (B,C,S)
__global__ void transpose_out_kernel(const float* __restrict__ src,
                                     float* __restrict__ out, int S, int C,
                                     int total) {
  int t = blockIdx.x * blockDim.x + threadIdx.x;
  if (t >= total) return;            // total = B*C*S, t indexes output
  int s = t % S;
  int c = (t / S) % C;
  int b = t / (S * C);
  out[t] = src[((size_t)b * S + s) * C + c];
}

// ===========================================================================
// Host orchestration
// ===========================================================================
extern "C" void kernel_launch(void* const* d_in, const int* in_sizes, int n_in,
                              void* d_out, int out_size, void* d_ws,
                              size_t ws_size, hipStream_t stream) {
  (void)in_sizes; (void)n_in; (void)out_size; (void)ws_size;
  const int B = 8, N0 = 4096;
  const int KNB = 32;                 // nsample
  const float* pc = (const float*)d_in[0];

  struct Layer {
    const float *W, *b, *g, *bt;
    int cin, cout;
  };
  int p = 1;
  auto take = [&](int cin, int cout, bool bn) {
    Layer l;
    l.W = (const float*)d_in[p++];
    l.b = (const float*)d_in[p++];
    l.g = bn ? (const float*)d_in[p++] : nullptr;
    l.bt = bn ? (const float*)d_in[p++] : nullptr;
    l.cin = cin;
    l.cout = cout;
    return l;
  };
  const int sa_cin[4] = {12, 67, 131, 259};
  const int sa_ch[4][3] = {{32, 32, 64}, {64, 64, 128}, {128, 128, 256}, {256, 256, 512}};
  Layer sa[4][3];
  for (int s = 0; s < 4; ++s) {
    int c = sa_cin[s];
    for (int i = 0; i < 3; ++i) { sa[s][i] = take(c, sa_ch[s][i], true); c = sa_ch[s][i]; }
  }
  const int fp_cin[4] = {768, 384, 320, 128};
  const int fp_nl[4] = {2, 2, 2, 3};
  const int fp_ch[4][3] = {{256, 256, 0}, {256, 256, 0}, {256, 128, 0}, {128, 128, 128}};
  Layer fp[4][3];
  for (int s = 0; s < 4; ++s) {
    int c = fp_cin[s];
    for (int i = 0; i < fp_nl[s]; ++i) { fp[s][i] = take(c, fp_ch[s][i], true); c = fp_ch[s][i]; }
  }
  Layer head1 = take(128, 128, true);
  Layer head2 = take(128, 13, false);

  // ---- workspace layout (bump allocator, 256B aligned) ----
  char* base = (char*)d_ws;
  size_t off = 0;
  auto alloc = [&](size_t bytes) -> void* {
    void* r = base + off;
    off += (bytes + 255) & ~(size_t)255;
    return r;
  };
  const int npts[5] = {4096, 1024, 256, 64, 16};
  const int pch[5] = {9, 64, 128, 256, 512};
  float* x0 = (float*)alloc((size_t)B * N0 * 9 * 4);
  float* xyz0 = (float*)alloc((size_t)B * N0 * 3 * 4);
  float* lx[5];
  float* lp[5];
  lx[0] = xyz0;
  lp[0] = x0;
  for (int s = 1; s <= 4; ++s) {
    lx[s] = (float*)alloc((size_t)B * npts[s] * 3 * 4);
    lp[s] = (float*)alloc((size_t)B * npts[s] * pch[s] * 4);
  }
  const int fpo_ch[4] = {256, 256, 128, 128};   // p3,p2,p1,p0
  float* fpout[4];
  for (int s = 0; s < 4; ++s)
    fpout[s] = (float*)alloc((size_t)B * npts[3 - s] * fpo_ch[s] * 4);
  float* scale = (float*)alloc(512 * 4);
  float* shift = (float*)alloc(512 * 4);
  int* fpsidx = (int*)alloc((size_t)B * 1024 * 4);
  int* gi = (int*)alloc((size_t)B * 1024 * KNB * 4);
  float* PA = (float*)alloc((size_t)8388608 * 4);    // 32 MB ping
  float* PB = (float*)alloc((size_t)16777216 * 4);   // 64 MB pong

  auto cdiv = [](long long a, long long b) { return (int)((a + b - 1) / b); };
  const int TPB = 256;

  auto gemm = [&](const float* X, const Layer& l, float* Y, int rows) {
    int tiles = (rows >> 4) * ((l.cout + 31) >> 5);   // 16x32 strips
    int blocks = cdiv(tiles, TPB / 32);
    wmma_gemm_bias_kernel<<<blocks, TPB, 0, stream>>>(X, l.W, l.b, Y, rows,
                                                      l.cin, l.cout);
  };
  auto bn = [&](float* Y, const Layer& l, int rows) {
    bn_stats_kernel<<<l.cout, TPB, 0, stream>>>(Y, l.g, l.bt, scale, shift,
                                                rows, l.cout);
    size_t total = (size_t)rows * l.cout;
    bn_apply_kernel<<<cdiv((long long)total, TPB), TPB, 0, stream>>>(
        Y, scale, shift, total, l.cout);
  };
  auto run_mlp = [&](const float* Xin, int rows, Layer* ls, int nl,
                     float* outFinal) -> float* {
    const float* cur = Xin;
    float* out = nullptr;
    for (int i = 0; i < nl; ++i) {
      out = (i == nl - 1) ? outFinal : (((i & 1) == 0) ? PB : PA);
      gemm(cur, ls[i], out, rows);
      bn(out, ls[i], rows);
      cur = out;
    }
    return out;
  };

  // ---- prep ----
  {
    int total = B * 9 * N0;
    transpose_pc_kernel<<<cdiv(total, TPB), TPB, 0, stream>>>(pc, x0, xyz0, 9,
                                                              N0, total);
  }

  // ---- set abstraction stages ----
  const float radius[4] = {0.1f, 0.2f, 0.4f, 0.8f};
  for (int s = 0; s < 4; ++s) {
    int Nin = npts[s], Sout = npts[s + 1], Cp = pch[s];
    const float* xyzin = lx[s];
    const float* ptsin = lp[s];
    size_t smem = (size_t)Nin * 4 + (size_t)TPB * 8;
    fps_kernel<<<B, TPB, smem, stream>>>(xyzin, fpsidx, Nin, Sout);
    int bs = B * Sout;
    gather_xyz2_kernel<<<cdiv(bs, TPB), TPB, 0, stream>>>(xyzin, fpsidx,
                                                          lx[s + 1], Nin, Sout,
                                                          bs);
    ball_query_kernel<<<cdiv(bs, TPB), TPB, 0, stream>>>(
        xyzin, lx[s + 1], gi, Nin, Sout, radius[s] * radius[s], KNB, bs);
    int rows = bs * KNB;
    group_feat_kernel<<<cdiv(rows, TPB), TPB, 0, stream>>>(
        xyzin, lx[s + 1], ptsin, gi, PA, Nin, Sout, KNB, Cp, rows);
    float* feat = run_mlp(PA, rows, sa[s], 3, PB);
    int ptot = bs * pch[s + 1];
    maxpool_kernel<<<cdiv(ptot, TPB), TPB, 0, stream>>>(feat, lp[s + 1], KNB,
                                                        pch[s + 1], ptot);
  }

  // ---- feature propagation stages ----
  for (int s = 0; s < 4; ++s) {
    int lvl1 = 3 - s, lvl2 = 4 - s;
    int n1 = npts[lvl1], n2 = npts[lvl2];
    int C1 = (s < 3) ? pch[lvl1] : 0;
    const float* pts2 = (s == 0) ? lp[4] : fpout[s - 1];
    int C2 = (s == 0) ? 512 : fpo_ch[s - 1];
    int Ctot = fp_cin[s];
    int rows = B * n1;
    if (C1 > 0) {
      int total = rows * C1;
      copy_cols_kernel<<<cdiv(total, TPB), TPB, 0, stream>>>(lp[lvl1], PA, C1,
                                                             Ctot, total);
    }
    fp_interp_kernel<<<cdiv(rows, TPB), TPB, 0, stream>>>(
        lx[lvl1], lx[lvl2], pts2, PA, n1, n2, C2, Ctot, C1, rows);
    run_mlp(PA, rows, fp[s], fp_nl[s], fpout[s]);
  }

  // ---- heads ----
  int rowsH = B * N0;                       // 32768
  run_mlp(fpout[3], rowsH, &head1, 1, PB);  // head1 + BN + relu -> PB
  gemm(PB, head2, PA, rowsH);               // head2: linear + bias -> PA
  log_softmax_kernel<<<cdiv(rowsH, TPB), TPB, 0, stream>>>(PA, (float*)d_out,
                                                           13, rowsH);
  // second output: l4p (8,16,512) -> (8,512,16)
  {
    int total = B * 512 * 16;
    transpose_out_kernel<<<cdiv(total, TPB), TPB, 0, stream>>>(
        lp[4], (float*)d_out + (size_t)rowsH * 13, 16, 512, total);
  }
}